// GraphConvolution_Node_11562051961573
// MI455X (gfx1250) — compile-verified
//
#include <hip/hip_runtime.h>
#include <math.h>

// CDNA5 (gfx1250) GraphConvolution:
//   agg = segment_sum(edge_val * x[edge_src], edge_dst)   (fp32 atomics, L2-resident)
//   out = agg @ W                                          (V_WMMA_F32_16X16X4_F32)
//   out = relu(out / max(||out||_2, 1e-12))                (wave32 shuffle reduction)

typedef float v2f __attribute__((ext_vector_type(2)));
typedef float v8f __attribute__((ext_vector_type(8)));

#define D 64

// ---------------- zero the accumulator workspace ----------------
__global__ void zero_kernel(float4* __restrict__ p, long n4) {
  long i = (long)blockIdx.x * blockDim.x + threadIdx.x;
  if (i < n4) p[i] = make_float4(0.f, 0.f, 0.f, 0.f);
}

// ---------------- edge scatter: agg[dst] += val * x[src] ----------------
// 16 lanes per edge, each lane owns a float4 chunk of the 64-float row.
__global__ void scatter_kernel(const float* __restrict__ x,
                               const float* __restrict__ ev,
                               const int* __restrict__ src,
                               const int* __restrict__ dst,
                               float* __restrict__ agg, int n_edges) {
  long gid = (long)blockIdx.x * blockDim.x + threadIdx.x;
  int e = (int)(gid >> 4);
  if (e >= n_edges) return;
  int c = ((int)gid & 15) << 2;
  int s = src[e];
  int d = dst[e];
  float v = ev[e];
  const float4 xv = *(const float4*)(x + (long)s * D + c);
  float* a = agg + (long)d * D + c;
  atomicAdd(a + 0, xv.x * v);
  atomicAdd(a + 1, xv.y * v);
  atomicAdd(a + 2, xv.z * v);
  atomicAdd(a + 3, xv.w * v);
}

// ---------------- out = agg @ W via fp32 WMMA ----------------
// One wave computes a 16x64 tile: 4 accumulators (N-tiles of 16) x 16 K-steps.
__global__ void gemm_wmma_kernel(const float* __restrict__ agg,
                                 const float* __restrict__ W,
                                 float* __restrict__ out, int n_nodes) {
  const int lane = threadIdx.x & 31;
  const int wave = threadIdx.x >> 5;
  const int row0 = (blockIdx.x * 8 + wave) * 16;
  if (row0 >= n_nodes) return;  // wave-uniform: EXEC stays all-ones below

  const int m    = lane & 15;   // A: row M; B/C/D: column N (within tile)
  const int half = lane >> 4;   // 0: K={0,1} / rows 0-7; 1: K={2,3} / rows 8-15
  const int klo  = half * 2;

  int rowA = row0 + m;
  if (rowA >= n_nodes) rowA = n_nodes - 1;  // clamp loads only; stores masked later
  const float* __restrict__ arow = agg + (long)rowA * D;

  v8f acc0 = {}, acc1 = {}, acc2 = {}, acc3 = {};

#pragma unroll
  for (int kt = 0; kt < 16; ++kt) {
    const int k = kt * 4 + klo;
    // A fragment 16x4: lane holds A[m][k], A[m][k+1]
    v2f a;
    a.x = arow[k];
    a.y = arow[k + 1];
    // B fragments 4x16 per N-tile: lane holds W[k][n], W[k+1][n]
    const float* __restrict__ b0 = W + (long)k * D + m;
    const float* __restrict__ b1 = b0 + D;
    v2f bA; bA.x = b0[0];  bA.y = b1[0];
    v2f bB; bB.x = b0[16]; bB.y = b1[16];
    v2f bC; bC.x = b0[32]; bC.y = b1[32];
    v2f bD; bD.x = b0[48]; bD.y = b1[48];

    acc0 = __builtin_amdgcn_wmma_f32_16x16x4_f32(false, a, false, bA, (short)0, acc0, false, false);
    acc1 = __builtin_amdgcn_wmma_f32_16x16x4_f32(false, a, false, bB, (short)0, acc1, false, false);
    acc2 = __builtin_amdgcn_wmma_f32_16x16x4_f32(false, a, false, bC, (short)0, acc2, false, false);
    acc3 = __builtin_amdgcn_wmma_f32_16x16x4_f32(false, a, false, bD, (short)0, acc3, false, false);
  }

  // C/D layout: VGPR r -> row (r + half*8), column = m (+ N-tile offset)
#pragma unroll
  for (int r = 0; r < 8; ++r) {
    const int row = row0 + r + half * 8;
    if (row < n_nodes) {
      float* __restrict__ o = out + (long)row * D + m;
      o[0]  = acc0[r];
      o[16] = acc1[r];
      o[32] = acc2[r];
      o[48] = acc3[r];
    }
  }
}

// ---------------- row-wise L2 normalize + ReLU (in place) ----------------
// One wave32 per node; lane owns columns {lane, lane+32}.
__global__ void norm_relu_kernel(float* __restrict__ out, int n_nodes) {
  const int node = blockIdx.x * (blockDim.x >> 5) + (threadIdx.x >> 5);
  const int lane = threadIdx.x & 31;
  if (node >= n_nodes) return;
  float* __restrict__ row = out + (long)node * D;
  float a = row[lane];
  float b = row[lane + 32];
  float ss = a * a + b * b;
#pragma unroll
  for (int off = 16; off > 0; off >>= 1)
    ss += __shfl_xor(ss, off, 32);
  const float inv = 1.0f / fmaxf(sqrtf(ss), 1e-12f);
  row[lane]      = fmaxf(a * inv, 0.0f);
  row[lane + 32] = fmaxf(b * inv, 0.0f);
}

extern "C" void kernel_launch(void* const* d_in, const int* in_sizes, int n_in,
                              void* d_out, int out_size, void* d_ws, size_t ws_size,
                              hipStream_t stream) {
  const float* x   = (const float*)d_in[0];  // [N, 64]
  const float* ev  = (const float*)d_in[1];  // [E]
  const float* W   = (const float*)d_in[2];  // [64, 64]
  const int*   src = (const int*)d_in[3];    // [E]
  const int*   dst = (const int*)d_in[4];    // [E]
  float* out = (float*)d_out;                // [N, 64]
  float* agg = (float*)d_ws;                 // [N, 64] scratch accumulator

  const int n_nodes = in_sizes[0] / D;
  const int n_edges = in_sizes[1];

  // 1) agg = 0
  const long n4 = (long)n_nodes * D / 4;
  zero_kernel<<<(unsigned)((n4 + 255) / 256), 256, 0, stream>>>((float4*)agg, n4);

  // 2) scatter-add messages (16 lanes per edge)
  const long sthreads = (long)n_edges * 16;
  scatter_kernel<<<(unsigned)((sthreads + 255) / 256), 256, 0, stream>>>(
      x, ev, src, dst, agg, n_edges);

  // 3) out = agg @ W  (8 waves/block, one 16-row tile per wave)
  const int tiles  = (n_nodes + 15) / 16;
  const int blocks = (tiles + 7) / 8;
  gemm_wmma_kernel<<<blocks, 256, 0, stream>>>(agg, W, out, n_nodes);

  // 4) normalize + relu (8 waves/block, one node per wave)
  const int nb = (n_nodes + 7) / 8;
  norm_relu_kernel<<<nb, 256, 0, stream>>>(out, n_nodes);
}